// EmbeddingGraph_27951647163111
// MI455X (gfx1250) — compile-verified
//
#include <hip/hip_runtime.h>
#include <hip/hip_bf16.h>

typedef __attribute__((ext_vector_type(2))) float v2f;
typedef __attribute__((ext_vector_type(8))) float v8f;

#define N_NODES 100000
#define HIDDEN  128
#define N_CLASS 10
#define N_EDGES 1600000

// LDS padding strides (in floats) chosen for conflict-free access on 64 banks:
//  X strip: stride 132 -> A-frag reads (16 rows, same col) hit banks 4m+k, all distinct.
//  W tile : stride 136 -> B-frag reads (rows k,k+1 for lanes 0-15; k+2,k+3 for 16-31)
//           land on disjoint 16-bank groups (offset 2*136 = 272 = 16 mod 64).
#define XS 132
#define WS 136

// ---------------- degree / normalization ----------------

__global__ void deg_init_kernel(float* __restrict__ deg, int n) {
    int i = blockIdx.x * blockDim.x + threadIdx.x;
    if (i < n) deg[i] = 1.0f;  // self-loop contribution
}

__global__ void deg_accum_kernel(const int* __restrict__ dst, float* __restrict__ deg, int e) {
    int i = blockIdx.x * blockDim.x + threadIdx.x;
    if (i < e) atomicAdd(&deg[dst[i]], 1.0f);
}

__global__ void inv_sqrt_kernel(float* __restrict__ deg, int n) {
    int i = blockIdx.x * blockDim.x + threadIdx.x;
    if (i < n) deg[i] = rsqrtf(deg[i]);  // deg >= 1 always
}

// ---------------- dense 128x128 GEMM via fp32 WMMA ----------------
// One block = 8 waves = one 16-row strip of X. Wave w computes the 16x16
// output tile at columns [16w, 16w+16). 32 x V_WMMA_F32_16X16X4_F32 per tile.

__global__ void __launch_bounds__(256) gemm128_wmma_kernel(
        const float* __restrict__ X,   // [N,128] row-major
        const float* __restrict__ W,   // [128,128] row-major
        float* __restrict__ Hout) {    // [N,128] row-major
    __shared__ float Xl[16 * XS];
    __shared__ float Wl[HIDDEN * WS];

    const int tid     = threadIdx.x;
    const int lane    = tid & 31;
    const int wave    = tid >> 5;
    const int rowTile = blockIdx.x * 16;

    // Stage W (64KB) into LDS with padded stride.
    for (int idx = tid; idx < HIDDEN * HIDDEN; idx += 256) {
        int r = idx >> 7, c = idx & 127;
        Wl[r * WS + c] = W[idx];
    }
    // Stage the 16x128 X strip (shared by all 8 waves).
    for (int idx = tid; idx < 16 * HIDDEN; idx += 256) {
        int r = idx >> 7, c = idx & 127;
        Xl[r * XS + c] = X[(size_t)(rowTile + r) * HIDDEN + c];
    }
    __syncthreads();

    const int m    = lane & 15;      // matrix row (A) / col (B,C)
    const int half = lane >> 4;      // lanes 16-31 carry K+2 / M+8 per ISA layout
    const int kh   = half * 2;
    const int col  = wave * 16 + m;

    v8f acc = {};
    #pragma unroll 4
    for (int k0 = 0; k0 < HIDDEN; k0 += 4) {
        v2f a, b;
        a.x = Xl[m * XS + k0 + kh];            // A[m, k0+kh]
        a.y = Xl[m * XS + k0 + kh + 1];        // A[m, k0+kh+1]
        b.x = Wl[(k0 + kh) * WS + col];        // B[k0+kh,   col]
        b.y = Wl[(k0 + kh + 1) * WS + col];    // B[k0+kh+1, col]
        acc = __builtin_amdgcn_wmma_f32_16x16x4_f32(
            /*neg_a=*/false, a, /*neg_b=*/false, b,
            /*c_mod=*/(short)0, acc, /*reuse_a=*/false, /*reuse_b=*/false);
    }

    // C/D layout: VGPR v holds row (v + 8*half), lane column = col.
    const int rbase = rowTile + half * 8;
    #pragma unroll
    for (int v = 0; v < 8; ++v)
        Hout[(size_t)(rbase + v) * HIDDEN + col] = acc[v];
}

// ---------------- aggregation ----------------

// agg[i,:] = h[i,:] * inv[i]^2   (self-loop term; also initializes agg)
__global__ void agg_init_kernel(const float* __restrict__ h,
                                const float* __restrict__ inv,
                                float* __restrict__ agg, int total) {
    int idx = blockIdx.x * blockDim.x + threadIdx.x;
    if (idx >= total) return;
    int node = idx >> 7;
    float w = inv[node];
    agg[idx] = h[idx] * (w * w);
}

// One wave per edge: coalesced 512B row gather + coalesced f32 atomics.
__global__ void __launch_bounds__(256) edge_scatter_kernel(
        const int* __restrict__ src, const int* __restrict__ dst,
        const float* __restrict__ inv,
        const float* __restrict__ h, float* __restrict__ agg, int e) {
    int edge = blockIdx.x * 8 + (threadIdx.x >> 5);
    if (edge >= e) return;
    int lane = threadIdx.x & 31;
    int s = src[edge];
    int d = dst[edge];
    float norm = inv[s] * inv[d];
    float4 v = ((const float4*)(h + (size_t)s * HIDDEN))[lane];
    float* ap = agg + (size_t)d * HIDDEN + lane * 4;
    atomicAdd(ap + 0, v.x * norm);
    atomicAdd(ap + 1, v.y * norm);
    atomicAdd(ap + 2, v.z * norm);
    atomicAdd(ap + 3, v.w * norm);
}

// x = relu(agg + b), in place.
__global__ void relu_bias_kernel(float* __restrict__ x,
                                 const float* __restrict__ b, int total) {
    int idx = blockIdx.x * blockDim.x + threadIdx.x;
    if (idx >= total) return;
    float v = x[idx] + b[idx & 127];
    x[idx] = v > 0.0f ? v : 0.0f;
}

// ---------------- output projection [N,128] @ [128,10] + b ----------------

__global__ void out_proj_kernel(const float* __restrict__ x,
                                const float* __restrict__ Wo,
                                const float* __restrict__ bo,
                                float* __restrict__ out, int total) {
    int idx = blockIdx.x * blockDim.x + threadIdx.x;
    if (idx >= total) return;
    int i = idx / N_CLASS;
    int c = idx - i * N_CLASS;
    const float* xr = x + (size_t)i * HIDDEN;
    float s = bo[c];
    #pragma unroll 8
    for (int k = 0; k < HIDDEN; ++k)
        s = fmaf(xr[k], Wo[k * N_CLASS + c], s);
    out[idx] = s;
}

// ---------------- launcher ----------------

extern "C" void kernel_launch(void* const* d_in, const int* in_sizes, int n_in,
                              void* d_out, int out_size, void* d_ws, size_t ws_size,
                              hipStream_t stream) {
    const int*   edge_index = (const int*)d_in[0];     // [2, E]
    const int*   src   = edge_index;
    const int*   dst   = edge_index + N_EDGES;
    const float* emb   = (const float*)d_in[1];
    const float* W1    = (const float*)d_in[2];
    const float* b1    = (const float*)d_in[3];
    const float* W2    = (const float*)d_in[4];
    const float* b2    = (const float*)d_in[5];
    const float* W_out = (const float*)d_in[6];
    const float* b_out = (const float*)d_in[7];
    float*       out   = (float*)d_out;

    // Workspace layout: deg/inv (N floats, 16B-aligned size) | bufA | bufB
    char*  ws   = (char*)d_ws;
    float* deg  = (float*)ws;
    float* bufA = (float*)(ws + (size_t)N_NODES * sizeof(float)); // 400000B, 16B aligned
    float* bufB = bufA + (size_t)N_NODES * HIDDEN;

    const int feat_total = N_NODES * HIDDEN;
    const int nb_nodes   = (N_NODES + 255) / 256;
    const int nb_edges   = (N_EDGES + 255) / 256;
    const int nb_feat    = (feat_total + 255) / 256;
    const int nb_scatter = (N_EDGES + 7) / 8;
    const int nb_tiles   = N_NODES / 16;               // 100000/16 = 6250 exact
    const int nb_out     = (N_NODES * N_CLASS + 255) / 256;

    // ---- normalization: deg -> inv_sqrt (stored in `deg`) ----
    deg_init_kernel <<<nb_nodes, 256, 0, stream>>>(deg, N_NODES);
    deg_accum_kernel<<<nb_edges, 256, 0, stream>>>(dst, deg, N_EDGES);
    inv_sqrt_kernel <<<nb_nodes, 256, 0, stream>>>(deg, N_NODES);

    // ---- layer 1: h1 = emb @ W1 ; agg ; x1 = relu(agg + b1) ----
    gemm128_wmma_kernel<<<nb_tiles, 256, 0, stream>>>(emb, W1, bufA);
    agg_init_kernel    <<<nb_feat, 256, 0, stream>>>(bufA, deg, bufB, feat_total);
    edge_scatter_kernel<<<nb_scatter, 256, 0, stream>>>(src, dst, deg, bufA, bufB, N_EDGES);
    relu_bias_kernel   <<<nb_feat, 256, 0, stream>>>(bufB, b1, feat_total);

    // ---- layer 2: h2 = x1 @ W2 ; agg ; x2 = relu(agg + b2) ----
    gemm128_wmma_kernel<<<nb_tiles, 256, 0, stream>>>(bufB, W2, bufA);
    agg_init_kernel    <<<nb_feat, 256, 0, stream>>>(bufA, deg, bufB, feat_total);
    edge_scatter_kernel<<<nb_scatter, 256, 0, stream>>>(src, dst, deg, bufA, bufB, N_EDGES);
    relu_bias_kernel   <<<nb_feat, 256, 0, stream>>>(bufB, b2, feat_total);

    // ---- output projection ----
    out_proj_kernel<<<nb_out, 256, 0, stream>>>(bufB, W_out, b_out, out, N_NODES * N_CLASS);
}